// MCLSTM_65386582114595
// MI455X (gfx1250) — compile-verified
//
#include <hip/hip_runtime.h>
#include <hip/hip_bf16.h>

typedef __attribute__((ext_vector_type(2))) float v2f;
typedef __attribute__((ext_vector_type(8))) float v8f;

#define BT 1024      // batch
#define TT 366       // total timesteps
#define NS 365       // scan steps
#define CD 24        // C_DIM
#define KP 32        // padded K (29 data + bias col + 2 zero)
#define MT 16        // batch tile per workgroup
#define NZ 640       // padded z width (576 redis + 64 gate)
#define NRTILES 36   // redis N-tiles
#define NTILES 40    // total N-tiles (36 redis + 4 gate)
#define TPW 5        // tiles per wave (8 waves)

// Fetch one element of the combined, K-padded, bias-augmented weight matrix
// B[k][n] for WMMA B-operand. tile<36 -> W_redis region; tiles 36..39 -> gates.
__device__ __forceinline__ float fetchW(int tile, int n, int k,
    const float* Wred, const float* bred,
    const float* Wpart, const float* bpart,
    const float* Wcons, const float* bcons,
    const float* Wass,  const float* bass)
{
  if (tile < NRTILES) {
    if (k < 29)  return Wred[n * 29 + k];
    if (k == 29) return bred[n];
    return 0.f;
  }
  const int ng = n - 576;
  if (ng < 24) { if (k < 29) return Wpart[ng * 29 + k]; if (k == 29) return bpart[ng]; return 0.f; }
  if (ng < 48) { const int m = ng - 24;
                 if (k < 29) return Wcons[m * 29 + k];  if (k == 29) return bcons[m];  return 0.f; }
  if (ng == 48){ if (k < 29) return Wass[k];            if (k == 29) return bass[0];   return 0.f; }
  return 0.f;
}

__global__ __launch_bounds__(256, 1)
void mclstm_scan_kernel(const float* __restrict__ X,    const float* __restrict__ ORY,
                        const float* __restrict__ Wred, const float* __restrict__ bred,
                        const float* __restrict__ Wpart,const float* __restrict__ bpart,
                        const float* __restrict__ Wcons,const float* __restrict__ bcons,
                        const float* __restrict__ Wass, const float* __restrict__ bass,
                        const float* __restrict__ c2a,  const float* __restrict__ g2y,
                        float* __restrict__ out)
{
  __shared__ __align__(16) float xm[MT][KP];     // masked x (redis input); cols 0..23 == full x
  __shared__ __align__(16) float xauxf[MT][8];   // full-x aux cols 24..31 only
  __shared__ __align__(16) float zb[MT][NZ];     // GEMM outputs z
  __shared__ float cbuf[MT][CD];                 // C_cell state
  __shared__ float cpb[MT][CD];                  // C' (pre-redistribution)
  __shared__ float coef[MT][CD];                 // softmax-scaled coefficients
  __shared__ float cpotS[MT];
  __shared__ float ncumS[MT];

  float* out_day = out;
  float* out_cc  = out + (size_t)BT * TT * 7;
  float* out_cv  = out + (size_t)BT * TT * 7 + (size_t)BT * NS * CD;

  const int tid  = threadIdx.x;
  const int lane = tid & 31;
  const int wave = tid >> 5;
  const int row0 = blockIdx.x * MT;

  // ---- Preload WMMA B-operands into registers (reused for all 365*2 GEMMs) ----
  // B layout (16x16x4 f32): VGPR0 = rows K=k0 (lanes 0-15) / K=k0+2 (lanes 16-31),
  // VGPR1 = K=k0+1 / K=k0+3 ; N = tile*16 + lane%16.
  v2f Breg[TPW][8];
#pragma unroll
  for (int u = 0; u < TPW; ++u) {
    const int tile = wave + 8 * u;
    const int n    = tile * 16 + (lane & 15);
    const int koff = (lane >> 4) << 1;
#pragma unroll
    for (int kk = 0; kk < 8; ++kk) {
      const int k0 = 4 * kk + koff;
      v2f b = { fetchW(tile, n, k0,     Wred, bred, Wpart, bpart, Wcons, bcons, Wass, bass),
                fetchW(tile, n, k0 + 1, Wred, bred, Wpart, bpart, Wcons, bcons, Wass, bass) };
      Breg[u][kk] = b;
    }
  }

  // ---- Init state + all_day[:,0,:] = ORY[:,0,:7] ----
  for (int idx = tid; idx < MT * CD; idx += 256) cbuf[idx / CD][idx % CD] = 0.f;
  if (tid < MT) ncumS[tid] = 0.f;
  for (int idx = tid; idx < MT * 7; idx += 256) {
    const int b = idx / 7, c = idx % 7;
    out_day[((size_t)(row0 + b) * TT) * 7 + c] = ORY[((size_t)(row0 + b) * TT) * 7 + c];
  }
  __syncthreads();

  for (int s = 0; s < NS; ++s) {
    // ---- build aux columns (masked aux in xm[.][24..31], full aux in xauxf) ----
    if (tid < MT) {
      const int gb = row0 + tid;
      const float dvs = ORY[((size_t)gb * TT + s) * 7];
      const float rad = X[((size_t)gb * TT + s) * 4 + 0];
      const float tmx = X[((size_t)gb * TT + s) * 4 + 1];
      const float tmn = X[((size_t)gb * TT + s) * 4 + 2];
      const float prc = X[((size_t)gb * TT + s) * 4 + 3];
      const float nc  = ncumS[tid] + prc; ncumS[tid] = nc;
      float ta = 0.5f * (tmx + tmn) * 50.f;
      ta = fminf(fmaxf(ta, 10.f), 40.f);
      const float eff = 0.54f - (ta - 10.f) * (1.f / 30.f) * (0.54f - 0.36f);
      cpotS[tid] = rad * (eff * (1.f / 3.6f) * (12.f / 44.f));
      xauxf[tid][0] = dvs; xauxf[tid][1] = rad; xauxf[tid][2] = tmx; xauxf[tid][3] = tmn;
      xauxf[tid][4] = nc;  xauxf[tid][5] = 1.f; xauxf[tid][6] = 0.f; xauxf[tid][7] = 0.f;
      xm[tid][24] = 0.f; xm[tid][25] = rad; xm[tid][26] = 0.f; xm[tid][27] = 0.f;
      xm[tid][28] = 0.f; xm[tid][29] = 1.f; xm[tid][30] = 0.f; xm[tid][31] = 0.f;
    }
    for (int idx = tid; idx < MT * CD; idx += 256) {
      const int b = idx / CD, j = idx % CD;
      xm[b][j] = cbuf[b][j];           // shared data columns (identical for masked/full)
    }
    __syncthreads();

    // ---- A operands: slices 0..5 shared; 6,7 differ (masked vs full aux) ----
    const int am   = lane & 15;
    const int koff = (lane >> 4) << 1;
    v2f A[8], Af6, Af7;
#pragma unroll
    for (int kk = 0; kk < 8; ++kk) A[kk] = *(const v2f*)&xm[am][4 * kk + koff];
    Af6 = *(const v2f*)&xauxf[am][0 + koff];
    Af7 = *(const v2f*)&xauxf[am][4 + koff];

    // ---- GEMM1: z = [x_masked @ Wred.T | x_full @ Wgates.T] (+bias via K=29 col) ----
#pragma unroll
    for (int u = 0; u < TPW; ++u) {
      const int tile = wave + 8 * u;
      const bool msk = tile < NRTILES;
      const v2f a6 = msk ? A[6] : Af6;
      const v2f a7 = msk ? A[7] : Af7;
      v8f acc = {};
#pragma unroll
      for (int kk = 0; kk < 6; ++kk)
        acc = __builtin_amdgcn_wmma_f32_16x16x4_f32(false, A[kk], false, Breg[u][kk],
                                                    (short)0, acc, false, false);
      acc = __builtin_amdgcn_wmma_f32_16x16x4_f32(false, a6, false, Breg[u][6],
                                                  (short)0, acc, false, false);
      acc = __builtin_amdgcn_wmma_f32_16x16x4_f32(false, a7, false, Breg[u][7],
                                                  (short)0, acc, false, false);
      const int n  = tile * 16 + (lane & 15);
      const int mb = (lane >> 4) * 8;
#pragma unroll
      for (int r = 0; r < 8; ++r) zb[mb + r][n] = acc[r];
    }
    __syncthreads();

    // ---- gates: assim/part/cons -> C' ----
    if (tid < MT) {
      const int r = tid;
      float mx = -1e30f;
      for (int j = 0; j < CD; ++j) mx = fmaxf(mx, zb[r][576 + j]);
      float p[CD]; float ssum = 0.f;
      for (int j = 0; j < CD; ++j) { p[j] = __expf(zb[r][576 + j] - mx); ssum += p[j]; }
      const float inv   = 1.f / ssum;
      const float assim = 1.f / (1.f + __expf(-zb[r][576 + 48]));
      const float cin   = assim * cpotS[r];
      for (int j = 0; j < CD; ++j) {
        const float cons = 1.f / (1.f + __expf(-zb[r][576 + 24 + j]));
        cpb[r][j] = (cbuf[r][j] + cin * p[j] * inv) * (1.f - cons);
      }
    }
    __syncthreads();

    // ---- redis row softmax -> exp in place + coef = C'/rowsum ----
    for (int idx = tid; idx < MT * CD; idx += 256) {
      const int b = idx / CD, i = idx % CD;
      float* zr = &zb[b][i * CD];
      float mx = -1e30f;
      for (int j = 0; j < CD; ++j) mx = fmaxf(mx, zr[j]);
      float sm = 0.f;
      for (int j = 0; j < CD; ++j) { const float e = __expf(zr[j] - mx); zr[j] = e; sm += e; }
      coef[b][i] = cpb[b][i] / sm;
    }
    __syncthreads();

    // ---- matvec: C_new[j] = sum_i coef[i]*exp(z[i,j]); store C_cell; seed x2_masked ----
    for (int idx = tid; idx < MT * CD; idx += 256) {
      const int b = idx / CD, j = idx % CD;
      float a = 0.f;
      for (int i = 0; i < CD; ++i) a += coef[b][i] * zb[b][i * CD + j];
      cbuf[b][j] = a;
      xm[b][j]   = a;   // aux cols 24..31 of xm unchanged within this step
      out_cc[(((size_t)(row0 + b)) * NS + s) * CD + j] = a;
    }
    __syncthreads();

    // ---- all_day diagnostics for t=s+1 ----
    if (tid < MT) {
      const int r = tid; const int gb = row0 + r;
      float pai = 0.f, lea = 0.f, ste = 0.f, gra = 0.f, yie = 0.f;
      for (int j = 0; j < CD; ++j) pai += fabsf(cbuf[r][j] * c2a[j]);
      for (int j = 0; j < 8;  ++j) lea += cbuf[r][j];
      for (int j = 8; j < 16; ++j) ste += cbuf[r][j];
      for (int j = 16; j < 24; ++j) { gra += cbuf[r][j]; yie += fabsf(cbuf[r][j] * g2y[j - 16]); }
      lea *= (1.f / 0.419f); ste *= (1.f / 0.431f); gra *= (1.f / 0.487f); yie *= (1.f / 0.487f);
      float* o = &out_day[((size_t)gb * TT + (s + 1)) * 7];
      o[0] = ORY[((size_t)gb * TT + (s + 1)) * 7];
      o[1] = pai; o[2] = lea; o[3] = ste; o[4] = gra; o[5] = lea + ste + gra; o[6] = yie;
    }

    // ---- GEMM2: z2 = x2_masked @ Wred.T (+bias); reuse aux slices A[6],A[7] ----
    {
      v2f A2[6];
#pragma unroll
      for (int kk = 0; kk < 6; ++kk) A2[kk] = *(const v2f*)&xm[am][4 * kk + koff];
#pragma unroll
      for (int u = 0; u < TPW; ++u) {
        const int tile = wave + 8 * u;
        if (tile < NRTILES) {        // wave-uniform branch; EXEC stays full for WMMA
          v8f acc = {};
#pragma unroll
          for (int kk = 0; kk < 6; ++kk)
            acc = __builtin_amdgcn_wmma_f32_16x16x4_f32(false, A2[kk], false, Breg[u][kk],
                                                        (short)0, acc, false, false);
          acc = __builtin_amdgcn_wmma_f32_16x16x4_f32(false, A[6], false, Breg[u][6],
                                                      (short)0, acc, false, false);
          acc = __builtin_amdgcn_wmma_f32_16x16x4_f32(false, A[7], false, Breg[u][7],
                                                      (short)0, acc, false, false);
          const int n  = tile * 16 + (lane & 15);
          const int mb = (lane >> 4) * 8;
#pragma unroll
          for (int r = 0; r < 8; ++r) zb[mb + r][n] = acc[r];
        }
      }
    }
    __syncthreads();

    // ---- redis2 softmax coef2 = C/rowsum ----
    for (int idx = tid; idx < MT * CD; idx += 256) {
      const int b = idx / CD, i = idx % CD;
      float* zr = &zb[b][i * CD];
      float mx = -1e30f;
      for (int j = 0; j < CD; ++j) mx = fmaxf(mx, zr[j]);
      float sm = 0.f;
      for (int j = 0; j < CD; ++j) { const float e = __expf(zr[j] - mx); zr[j] = e; sm += e; }
      coef[b][i] = cbuf[b][i] / sm;
    }
    __syncthreads();

    // ---- matvec -> C_conv; store ----
    for (int idx = tid; idx < MT * CD; idx += 256) {
      const int b = idx / CD, j = idx % CD;
      float a = 0.f;
      for (int i = 0; i < CD; ++i) a += coef[b][i] * zb[b][i * CD + j];
      out_cv[(((size_t)(row0 + b)) * NS + s) * CD + j] = a;
    }
    __syncthreads();
  }
}

extern "C" void kernel_launch(void* const* d_in, const int* in_sizes, int n_in,
                              void* d_out, int out_size, void* d_ws, size_t ws_size,
                              hipStream_t stream) {
  const float* X     = (const float*)d_in[0];
  const float* ORY   = (const float*)d_in[1];
  const float* Wred  = (const float*)d_in[2];
  const float* bred  = (const float*)d_in[3];
  const float* Wpart = (const float*)d_in[4];
  const float* bpart = (const float*)d_in[5];
  const float* Wcons = (const float*)d_in[6];
  const float* bcons = (const float*)d_in[7];
  const float* Wass  = (const float*)d_in[8];
  const float* bass  = (const float*)d_in[9];
  const float* c2a   = (const float*)d_in[10];
  const float* g2y   = (const float*)d_in[11];
  float* out = (float*)d_out;

  dim3 grid(BT / MT);   // 64 workgroups, each owns 16 batch rows for all 365 steps
  dim3 block(256);      // 8 waves: 5 WMMA N-tiles each
  mclstm_scan_kernel<<<grid, block, 0, stream>>>(X, ORY, Wred, bred, Wpart, bpart,
                                                 Wcons, bcons, Wass, bass, c2a, g2y, out);
}